// NN_29102698397696
// MI455X (gfx1250) — compile-verified
//
#include <hip/hip_runtime.h>
#include <hip/hip_bf16.h>

// Problem constants (match reference).
#define BATCH     64
#define NUM_GENES 20000
#define HID       2048
#define GMAX      256

typedef __attribute__((ext_vector_type(2))) float v2f;
typedef __attribute__((ext_vector_type(8))) float v8f;

// ---------------------------------------------------------------------------
// Kernel 1: transpose x [64, 20000] -> xT [20000, 64] so that the ragged
// gather becomes contiguous 256B rows (one row = all 64 batches of one gene).
// ---------------------------------------------------------------------------
__global__ void xpose_kernel(const float* __restrict__ x, float* __restrict__ xT) {
    __shared__ float tile[64][65];            // +1 pad: conflict-free transpose
    const int t  = threadIdx.x;               // 0..255
    const int g0 = blockIdx.x * 64;
    const int c  = t & 63;                    // fast-varying lane index
    const int r4 = t >> 6;                    // 0..3
#pragma unroll
    for (int i = 0; i < 16; ++i) {
        const int b = i * 4 + r4;             // batch row 0..63
        const int g = g0 + c;                 // gene column (coalesced read)
        tile[b][c] = (g < NUM_GENES) ? x[b * NUM_GENES + g] : 0.0f;
    }
    __syncthreads();
#pragma unroll
    for (int i = 0; i < 16; ++i) {
        const int gr = i * 4 + r4;            // gene offset within tile
        const int g  = g0 + gr;
        if (g < NUM_GENES)
            xT[g * BATCH + c] = tile[c][gr];  // coalesced write along batch
    }
}

// ---------------------------------------------------------------------------
// Kernel 2: one workgroup per gene-group h.
//   hiddenR[h][b] = relu( b1[h] + sum_g (W1*mask)[h,g] * xT[idx[h,g]][b] )
// 256 threads = (4 g-slices) x (64 batches). Each wave's 32 lanes read 32
// consecutive floats of an xT row -> one fully-utilized 128B line per load.
// ---------------------------------------------------------------------------
__global__ void group_dot_kernel(const float* __restrict__ xT,
                                 const int*   __restrict__ gidx,
                                 const float* __restrict__ gmask,
                                 const float* __restrict__ W1,
                                 const float* __restrict__ b1,
                                 float*       __restrict__ hiddenR) { // [HID][64]
    const int h = blockIdx.x;
    __shared__ int   s_idx[GMAX];             // premultiplied by BATCH
    __shared__ float s_w[GMAX];               // W1 * mask, premultiplied once
    __shared__ float s_part[256];

    const int t = threadIdx.x;
    s_idx[t] = gidx[h * GMAX + t] * BATCH;
    s_w[t]   = W1[h * GMAX + t] * gmask[h * GMAX + t];
    __syncthreads();

    const int b  = t & 63;                    // batch owned by this thread
    const int gs = t >> 6;                    // g-slice 0..3
    float a0 = 0.f, a1 = 0.f, a2 = 0.f, a3 = 0.f;  // 4-way ILP accumulators
#pragma unroll 4
    for (int g = gs; g < GMAX; g += 16) {
        a0 = fmaf(s_w[g],      xT[s_idx[g]      + b], a0);
        a1 = fmaf(s_w[g + 4],  xT[s_idx[g + 4]  + b], a1);
        a2 = fmaf(s_w[g + 8],  xT[s_idx[g + 8]  + b], a2);
        a3 = fmaf(s_w[g + 12], xT[s_idx[g + 12] + b], a3);
    }
    s_part[t] = (a0 + a1) + (a2 + a3);
    __syncthreads();

    if (t < 64) {
        float v = s_part[t] + s_part[t + 64] + s_part[t + 128] + s_part[t + 192] + b1[h];
        hiddenR[h * BATCH + t] = fmaxf(v, 0.0f);   // fused bias + ReLU, coalesced
    }
}

// ---------------------------------------------------------------------------
// Kernel 3: out[64,2] = hiddenR^T[64,2048] @ W2[2048,2] + b2 via
// V_WMMA_F32_16X16X4_F32. 4 waves, one 16-row M-tile each; N padded to 16
// with a branchless column mask so EXEC stays all-ones through the WMMAs.
// A-layout (ISA 7.12.2, 32-bit A 16x4): lane<16 -> M=lane, K={0,1};
// lane>=16 -> M=lane-16, K={2,3}. B mirrored. D VGPR i -> rows m0+i / m0+i+8.
// ---------------------------------------------------------------------------
__global__ void fc2_wmma_kernel(const float* __restrict__ hR,  // [HID][64]
                                const float* __restrict__ W2,  // [HID][2]
                                const float* __restrict__ b2,  // [2]
                                float*       __restrict__ out) { // [64][2]
    const int lane = threadIdx.x & 31;
    const int wave = threadIdx.x >> 5;        // 0..3 -> M-tile
    const int m0   = wave << 4;
    const int m    = m0 + (lane & 15);
    const int n    = lane & 15;
    const int hi   = lane >> 4;               // 0 or 1 (K-half selector)
    const int kh   = hi << 1;                 // 0 or 2
    const float nmask = (n < 2) ? 1.0f : 0.0f;
    const int   ncol  = n & 1;

    v8f acc = {};
    for (int kg = 0; kg < HID; kg += 4) {
        const int k0 = kg + kh;
        v2f A, Bv;
        A.x  = hR[(k0    ) * BATCH + m];
        A.y  = hR[(k0 + 1) * BATCH + m];
        Bv.x = W2[(k0    ) * 2 + ncol] * nmask;   // zero for padded cols
        Bv.y = W2[(k0 + 1) * 2 + ncol] * nmask;
        acc = __builtin_amdgcn_wmma_f32_16x16x4_f32(
            /*neg_a=*/false, A, /*neg_b=*/false, Bv,
            /*c_mod=*/(short)0, acc, /*reuse_a=*/false, /*reuse_b=*/false);
    }

    if (n < 2) {
        const float bias = b2[n];
#pragma unroll
        for (int i = 0; i < 8; ++i) {
            const int row = m0 + i + (hi << 3);
            out[row * 2 + n] = acc[i] + bias;
        }
    }
}

// ---------------------------------------------------------------------------
extern "C" void kernel_launch(void* const* d_in, const int* in_sizes, int n_in,
                              void* d_out, int out_size, void* d_ws, size_t ws_size,
                              hipStream_t stream) {
    const float* x     = (const float*)d_in[0];   // [64, 20000]
    const int*   gidx  = (const int*)  d_in[1];   // [2048, 256]
    const float* gmask = (const float*)d_in[2];   // [2048, 256]
    const float* W1    = (const float*)d_in[3];   // [2048, 256]
    const float* b1    = (const float*)d_in[4];   // [2048]
    const float* W2    = (const float*)d_in[5];   // [2048, 2]
    const float* b2    = (const float*)d_in[6];   // [2]
    float*       out   = (float*)d_out;           // [64, 2]

    // Workspace layout: xT [20000][64] then hiddenR [2048][64].
    float* xT      = (float*)d_ws;
    float* hiddenR = xT + (size_t)NUM_GENES * BATCH;

    // 1) transpose x so gathers are contiguous per gene.
    const int n_tiles = (NUM_GENES + 63) / 64;    // 313
    xpose_kernel<<<n_tiles, 256, 0, stream>>>(x, xT);

    // 2) the heavy gather + per-group dot (one block per group).
    group_dot_kernel<<<HID, 256, 0, stream>>>(xT, gidx, gmask, W1, b1, hiddenR);

    // 3) fc2 via WMMA (single block, 4 waves).
    fc2_wmma_kernel<<<1, 128, 0, stream>>>(hiddenR, W2, b2, out);
}